// SlaterWF_27049704030839
// MI455X (gfx1250) — compile-verified
//
#include <hip/hip_runtime.h>
#include <stdint.h>

typedef __attribute__((ext_vector_type(2))) float v2f;
typedef __attribute__((ext_vector_type(8))) float v8f;

#define N_ELEC   16
#define N_MO     8
#define EMB      128
#define KCHUNKS  32          // 128 / 4
#define WAVES    2
#define BLOCK    (WAVES * 32)
#define CHUNK    8           // batch elements per wave
#define A_STRIDE 132         // padded aos row stride (floats) -> conflict-free ds_load_b64
#define XS_STRIDE 132        // per-element xs slot: up det at +0, down det at +65

// Issue 16 async global->LDS b128 copies (one aos row each), tracked by ASYNCcnt.
// GVS mode: 64-bit SGPR base plus 32-bit VGPR offset; LDS address taken from the VDST VGPR.
__device__ __forceinline__ void stage_async(float* lbuf, const float* gsrc, int lane) {
    uint64_t g   = (uint64_t)(uintptr_t)gsrc;                 // wave-uniform in fact
    uint32_t glo = __builtin_amdgcn_readfirstlane((uint32_t)g);
    uint32_t ghi = __builtin_amdgcn_readfirstlane((uint32_t)(g >> 32));
    uint64_t gu  = ((uint64_t)ghi << 32) | glo;               // provably uniform -> SGPR pair
    uint32_t lbase = (uint32_t)(uintptr_t)lbuf + (uint32_t)lane * 16u; // low 32 = LDS offset
    uint32_t goff  = (uint32_t)lane * 16u;
#pragma unroll
    for (int i = 0; i < N_ELEC; ++i) {
        asm volatile("global_load_async_to_lds_b128 %0, %1, %2"
                     :: "v"(lbase + (uint32_t)(i * A_STRIDE * 4)),
                        "v"(goff + (uint32_t)(i * EMB * 4)),
                        "s"(gu)
                     : "memory");
    }
}

// 8x8 determinant via LU with statically-indexed conditional-swap partial
// pivoting: all register indices are compile-time constants (no scratch).
__device__ __forceinline__ float det8(const float* __restrict__ m) {
    float a[64];
#pragma unroll
    for (int i = 0; i < 64; ++i) a[i] = m[i];
    float det = 1.0f;
#pragma unroll
    for (int k = 0; k < 8; ++k) {
#pragma unroll
        for (int i = k + 1; i < 8; ++i) {
            bool sw = __builtin_fabsf(a[i * 8 + k]) > __builtin_fabsf(a[k * 8 + k]);
#pragma unroll
            for (int j = k; j < 8; ++j) {       // cols < k never read again
                float x = a[k * 8 + j], y = a[i * 8 + j];
                a[k * 8 + j] = sw ? y : x;
                a[i * 8 + j] = sw ? x : y;
            }
            det = sw ? -det : det;
        }
        float piv = a[k * 8 + k];
        det *= piv;
        float inv = 1.0f / piv;
#pragma unroll
        for (int i = k + 1; i < 8; ++i) {
            float f = a[i * 8 + k] * inv;
#pragma unroll
            for (int j = k + 1; j < 8; ++j)
                a[i * 8 + j] = __builtin_fmaf(-f, a[k * 8 + j], a[i * 8 + j]);
        }
    }
    return det;
}

__global__ __launch_bounds__(BLOCK) void SlaterWF_27049704030839_kernel(
    const float* __restrict__ aos,   // [B, 16, 128]
    const float* __restrict__ W,     // [8, 128]
    float* __restrict__ out,         // [B]
    int batch) {
    __shared__ float sA[WAVES][2][N_ELEC * A_STRIDE];  // 33792 B: double-buffered tiles
    __shared__ float sXS[WAVES][CHUNK * XS_STRIDE];    //  8448 B: xs results / wave
    __shared__ v2f   sB[KCHUNKS * 32];                 //  8192 B: W in WMMA-B lane layout
    __shared__ float sDet[WAVES][16];                  //   128 B

    const int tid  = threadIdx.x;
    const int wid  = tid >> 5;
    const int lane = tid & 31;
    const int m    = lane & 15;   // M row (A) / N col (B,D)
    const int hi   = lane >> 4;   // K half-select

    // --- Precompute B fragments: B[k][n] = W[n][k] for n<8 else 0, in the
    // 16x16x4 f32 lane layout: lane(n,hi), vgpr v -> K = 4*kc + 2*hi + v.
    for (int idx = tid; idx < KCHUNKS * 32; idx += BLOCK) {
        int kc = idx >> 5, ln = idx & 31;
        int n = ln & 15, h = ln >> 4;
        int c0 = kc * 4 + h * 2;
        v2f val = {0.0f, 0.0f};
        if (n < N_MO) { val.x = W[n * EMB + c0]; val.y = W[n * EMB + c0 + 1]; }
        sB[idx] = val;
    }
    __syncthreads();

    // Hoist B fragments into registers (64 VGPRs, loop-invariant).
    v2f breg[KCHUNKS];
#pragma unroll
    for (int kc = 0; kc < KCHUNKS; ++kc) breg[kc] = sB[kc * 32 + lane];

    const int base = (blockIdx.x * WAVES + wid) * CHUNK;
    float* myXS = sXS[wid];

    // Pre-issue async staging of element 0 into buffer 0.
    if (base < batch)
        stage_async(sA[wid][0], aos + (size_t)base * (N_ELEC * EMB), lane);

    for (int s = 0; s < CHUNK; ++s) {
        const int b = base + s;
        if (b >= batch) break;                 // wave-uniform
        const float* gsrc = aos + (size_t)b * (N_ELEC * EMB);

        // Issue async staging of the NEXT element into the other buffer,
        // then wait for the current element's 16 copies (in-order completion).
        const bool next_valid = (s + 1 < CHUNK) && (b + 1 < batch);
        if (next_valid) {
            stage_async(sA[wid][(s + 1) & 1], gsrc + N_ELEC * EMB, lane);
            asm volatile("s_wait_asynccnt 0x10" ::: "memory");
        } else {
            asm volatile("s_wait_asynccnt 0x0" ::: "memory");
        }
        // Speculative prefetch two elements ahead (2 x 4KB, 128B-spaced lanes).
        if (b + 2 < batch) {
            __builtin_prefetch(gsrc + 2 * N_ELEC * EMB + lane * 32, 0, 1);
            __builtin_prefetch(gsrc + 2 * N_ELEC * EMB + 1024 + lane * 32, 0, 1);
        }

        // --- 16x8 <- 16x128 * 128x8 via 32 fp32 WMMAs, 2 accumulators.
        const float* myA = sA[wid][s & 1];
        v8f acc0 = {0.f, 0.f, 0.f, 0.f, 0.f, 0.f, 0.f, 0.f};
        v8f acc1 = acc0;
        const float* arow = myA + m * A_STRIDE + hi * 2;
#pragma unroll
        for (int kc = 0; kc < KCHUNKS; kc += 2) {
            v2f a0 = *reinterpret_cast<const v2f*>(arow + kc * 4);
            acc0 = __builtin_amdgcn_wmma_f32_16x16x4_f32(
                false, a0, false, breg[kc], (short)0, acc0, false, false);
            v2f a1 = *reinterpret_cast<const v2f*>(arow + (kc + 1) * 4);
            acc1 = __builtin_amdgcn_wmma_f32_16x16x4_f32(
                false, a1, false, breg[kc + 1], (short)0, acc1, false, false);
        }
#pragma unroll
        for (int v = 0; v < 8; ++v) acc0[v] += acc1[v];

        // --- Scatter xs (cols 0..7 only). D layout: vgpr v, lanes 0-15:
        // row v, col lane; lanes 16-31: row v+8, col lane-16.
        if (m < N_MO) {
            float* dst = myXS + s * XS_STRIDE + hi * 65 + m;
#pragma unroll
            for (int v = 0; v < 8; ++v) dst[v * 8] = acc0[v];
        }
        asm volatile("" ::: "memory");
    }

    asm volatile("" ::: "memory");
    // --- Determinant phase: lane (2e+sp) computes one 8x8 det of element e.
    if (lane < 16) {
        int e = lane >> 1, sp = lane & 1;
        if (base + e < batch)
            sDet[wid][lane] = det8(myXS + e * XS_STRIDE + sp * 65);
    }
    asm volatile("" ::: "memory");
    if (lane < CHUNK) {
        int b = base + lane;
        if (b < batch)
            out[b] = sDet[wid][2 * lane] * sDet[wid][2 * lane + 1] *
                     (1.0f / 40320.0f);   // (1/sqrt(8!))^2
    }
}

extern "C" void kernel_launch(void* const* d_in, const int* in_sizes, int n_in,
                              void* d_out, int out_size, void* d_ws, size_t ws_size,
                              hipStream_t stream) {
    const float* aos = (const float*)d_in[0];
    const float* W   = (const float*)d_in[1];
    float* out = (float*)d_out;
    const int batch  = in_sizes[0] / (N_ELEC * EMB);
    const int chunks = (batch + CHUNK - 1) / CHUNK;
    const int blocks = (chunks + WAVES - 1) / WAVES;
    SlaterWF_27049704030839_kernel<<<blocks, BLOCK, 0, stream>>>(aos, W, out, batch);
}